// Conv1d_64201171140713
// MI455X (gfx1250) — compile-verified
//
#include <hip/hip_runtime.h>

// Problem constants (from reference): x (16,32,32768) f32, w (1,32,32,3) f32
#define B_     16
#define CIN    32
#define LIN    32768
#define COUT   32
#define KW     3
#define LOUT   (LIN - KW + 1)     // 32766
#define KFLAT  (CIN * KW)         // 96

#define THREADS 256               // 8 wave32 waves
#define TILE_N  128               // output columns per block (16 per wave)
#define XROW    132               // padded LDS row: 128 + (KW-1) rounded to x4 (16B aligned)
#define XCHUNKS (XROW / 4)        // 33 float4 chunks per row

typedef __attribute__((ext_vector_type(2))) float v2f;
typedef __attribute__((ext_vector_type(8))) float v8f;

// Parameter type of __builtin_amdgcn_global_load_async_to_lds_b128 per hipcc
// diagnostic: 'int __attribute__((vector_size(16))) *' (generic AS pointer).
typedef int b128_t __attribute__((vector_size(4 * sizeof(int))));

#if __has_builtin(__builtin_amdgcn_global_load_async_to_lds_b128) && \
    __has_builtin(__builtin_amdgcn_s_wait_asynccnt)
#define HAVE_ASYNC_LDS 1
#else
#define HAVE_ASYNC_LDS 0
#endif

#define B128P(p) ((b128_t*)(p))

// One block: batch b = blockIdx.y, output columns [blockIdx.x*128, +128).
// Each wave computes a 32(C_out) x 16(columns) tile via fp32 WMMA.
// K dimension is permuted to kr' = ko*32 + ci (tap-major) so that per
// unrolled K-step the tap index is a compile-time constant and every LDS
// fragment load is base + immediate offset (no per-iteration VALU).
__global__ __launch_bounds__(THREADS)
void conv1d_wmma_f32(const float* __restrict__ x,
                     const float* __restrict__ w,
                     float* __restrict__ out)
{
    __shared__ __align__(16) float wlds[COUT * KFLAT];   // 12 KB, permuted (co, ko*32+ci)
    __shared__ __align__(16) float xlds[CIN][XROW];      // ~16.5 KB

    const int tid  = threadIdx.x;
    const int b    = blockIdx.y;
    const int nblk = blockIdx.x * TILE_N;

    const float* xb = x + (size_t)b * CIN * LIN;

    // ---- stage weights permuted: wlds[co*96 + ko*32 + ci] = w[co*96 + ci*3 + ko]
    for (int i = tid; i < COUT * KFLAT; i += THREADS) {
        const int co = i / KFLAT;
        const int r  = i - co * KFLAT;
        const int ci = r / KW;
        const int ko = r - ci * KW;
        wlds[co * KFLAT + ko * CIN + ci] = w[i];
    }

    // ---- stage x tile: 32 rows x 33 float4 chunks; clamp right edge into the row
#if HAVE_ASYNC_LDS
    for (int i = tid; i < CIN * XCHUNKS; i += THREADS) {
        const int row = i / XCHUNKS;
        const int chunk = i - row * XCHUNKS;
        int col = nblk + chunk * 4;
        if (col > LIN - 4) col = LIN - 4;   // only last block; slots unused for valid outputs
        __builtin_amdgcn_global_load_async_to_lds_b128(
            B128P(xb + (size_t)row * LIN + col),
            B128P(&xlds[row][chunk * 4]), 0, 0);
    }
    __builtin_amdgcn_s_wait_asynccnt(0);
#else
    for (int i = tid; i < CIN * XCHUNKS; i += THREADS) {
        const int row = i / XCHUNKS;
        const int chunk = i - row * XCHUNKS;
        int col = nblk + chunk * 4;
        if (col > LIN - 4) col = LIN - 4;
        *(float4*)&xlds[row][chunk * 4] = *(const float4*)(xb + (size_t)row * LIN + col);
    }
#endif
    __syncthreads();

    // ---------------- per-wave WMMA GEMM ----------------
    const int lane = tid & 31;
    const int wv   = tid >> 5;         // 0..7
    const int half = lane >> 4;        // selects K pair (A/B) and M+8 (C/D)
    const int nn   = lane & 15;        // column within tile / A-row index
    const int n0   = wv * 16;          // this wave's column offset in block tile

    // Per-lane bases; per-iteration offsets below are compile-time constants.
    const float* xf = &xlds[0][0];
    const int lb = n0 + nn + half * (2 * XROW);          // B-fragment base (float idx)
    const int la = nn * KFLAT + 2 * half;                // A-fragment base (float idx)

    v8f acc0 = {};                     // C_out 0..15
    v8f acc1 = {};                     // C_out 16..31

    #pragma unroll
    for (int kk = 0; kk < KFLAT / 4; ++kk) {
        const int ko = kk >> 3;                           // tap index: constant per kk
        const int C  = (4 * kk - 32 * ko) * XROW + ko;    // compile-time constant

        // B fragment (4x16): K = 4kk+2h+j maps to (ci = K-32ko, tap ko)
        v2f bf;
        bf.x = xf[lb + C];
        bf.y = xf[lb + C + XROW];

        // A fragments (16x4): row M=nn, two consecutive permuted-K values
        const v2f a0 = *(const v2f*)&wlds[la + 4 * kk];
        const v2f a1 = *(const v2f*)&wlds[la + 16 * KFLAT + 4 * kk];

        acc0 = __builtin_amdgcn_wmma_f32_16x16x4_f32(false, a0, false, bf,
                                                     (short)0, acc0, false, false);
        acc1 = __builtin_amdgcn_wmma_f32_16x16x4_f32(false, a1, false, bf,
                                                     (short)0, acc1, false, false);
    }

    // ---------------- store 32x16 tile (guard right edge) ----------------
    const int ng = nblk + n0 + nn;     // this lane's output column
    if (ng < LOUT) {
        // C/D layout: VGPR r holds M=r (lanes 0-15) / M=r+8 (lanes 16-31)
        float* ob = out + (size_t)b * COUT * LOUT + (size_t)(8 * half) * LOUT + ng;
        #pragma unroll
        for (int r = 0; r < 8; ++r) {
            ob[(size_t)r * LOUT]        = acc0[r];   // co = r + 8*half
            ob[(size_t)(r + 16) * LOUT] = acc1[r];   // co = r + 8*half + 16
        }
    }
}

extern "C" void kernel_launch(void* const* d_in, const int* in_sizes, int n_in,
                              void* d_out, int out_size, void* d_ws, size_t ws_size,
                              hipStream_t stream) {
    (void)in_sizes; (void)n_in; (void)out_size; (void)d_ws; (void)ws_size;
    const float* x = (const float*)d_in[0];   // (16, 32, 32768)
    const float* w = (const float*)d_in[1];   // (1, 32, 32, 3) -> flat (32, 96)
    float* out = (float*)d_out;               // (16, 32, 32766)

    dim3 grid((LOUT + TILE_N - 1) / TILE_N, B_);
    conv1d_wmma_f32<<<grid, THREADS, 0, stream>>>(x, w, out);
}